// Votenet_82471962018518
// MI455X (gfx1250) — compile-verified
//
#include <hip/hip_runtime.h>
#include <hip/hip_bf16.h>
#include <stdint.h>

#define B_ 4
#define N_ 16384
#define D_ 64
#define S_ 2048
#define K_ 32
#define R2_ (0.3f*0.3f)
#define EPS_ 1e-5f
#define NT_ (B_*S_*K_)   /* 262144 columns */

typedef float v2f __attribute__((ext_vector_type(2)));
typedef float v8f __attribute__((ext_vector_type(8)));

// ---------------- Farthest Point Sampling: one block per batch ----------------
// 1024 threads (32 waves, wave32). Each thread owns 16 points in registers.
__global__ __launch_bounds__(1024) void vn_fps(const float* __restrict__ xyz,
                                               const int* __restrict__ fps_start,
                                               int* __restrict__ fps_idx) {
  const int b = blockIdx.x;
  const int t = threadIdx.x;
  const float* xb = xyz + (size_t)b * 3 * N_;
  float xs[16], ys[16], zs[16], dm[16];
#pragma unroll
  for (int i = 0; i < 16; i++) {
    int n = i * 1024 + t;
    xs[i] = xb[n]; ys[i] = xb[N_ + n]; zs[i] = xb[2 * N_ + n];
    dm[i] = 1e10f;
  }
  __shared__ float sv[32];
  __shared__ int   si[32];
  __shared__ int   sfar;
  int far = fps_start[b];
  const int wv = t >> 5, ln = t & 31;
  int* out = fps_idx + b * S_;
  for (int s = 0; s < S_; s++) {
    if (t == 0) out[s] = far;
    float cx = xb[far], cy = xb[N_ + far], cz = xb[2 * N_ + far];
    float bv = -1.0f; int bi = 0x7fffffff;
#pragma unroll
    for (int i = 0; i < 16; i++) {
      float dx = xs[i] - cx, dy = ys[i] - cy, dz = zs[i] - cz;
      float d = dx * dx + dy * dy + dz * dz;
      dm[i] = fminf(dm[i], d);
      int n = i * 1024 + t;
      if (dm[i] > bv || (dm[i] == bv && n < bi)) { bv = dm[i]; bi = n; }
    }
#pragma unroll
    for (int off = 16; off; off >>= 1) {
      float ov = __shfl_xor(bv, off, 32);
      int   oi = __shfl_xor(bi, off, 32);
      if (ov > bv || (ov == bv && oi < bi)) { bv = ov; bi = oi; }
    }
    if (ln == 0) { sv[wv] = bv; si[wv] = bi; }
    __syncthreads();
    if (wv == 0) {
      bv = sv[ln]; bi = si[ln];
#pragma unroll
      for (int off = 16; off; off >>= 1) {
        float ov = __shfl_xor(bv, off, 32);
        int   oi = __shfl_xor(bi, off, 32);
        if (ov > bv || (ov == bv && oi < bi)) { bv = ov; bi = oi; }
      }
      if (ln == 0) sfar = bi;
    }
    __syncthreads();
    far = sfar;
  }
}

// -------- Ball query + gather + build X0[68, NT] + write new_xyz / seeds --------
// One wave per (b, s). Slot j (of K_=32) is held by lane j.
__global__ __launch_bounds__(256) void vn_ballgather(
    const float* __restrict__ xyz, const float* __restrict__ points,
    const int* __restrict__ seed_inds, const int* __restrict__ fps_idx,
    float* __restrict__ X0, float* __restrict__ out_nx, float* __restrict__ out_seed) {
  const int wid = blockIdx.x * 8 + (threadIdx.x >> 5);
  const int ln  = threadIdx.x & 31;
  const int b = wid / S_;
  const int s = wid % S_;
  const float* xb = xyz + (size_t)b * 3 * N_;
  const int far = fps_idx[b * S_ + s];
  const float cx = xb[far], cy = xb[N_ + far], cz = xb[2 * N_ + far];
  if (ln == 0) out_nx[(b * 3 + 0) * S_ + s] = cx;
  if (ln == 1) out_nx[(b * 3 + 1) * S_ + s] = cy;
  if (ln == 2) out_nx[(b * 3 + 2) * S_ + s] = cz;
  if (ln == 3) out_seed[b * S_ + s] = (float)seed_inds[(size_t)b * N_ + far];

  int sel = -1, cnt = 0;
  for (int base = 0; base < N_; base += 32) {
    int n = base + ln;
    float dx = xb[n] - cx, dy = xb[N_ + n] - cy, dz = xb[2 * N_ + n] - cz;
    float d = dx * dx + dy * dy + dz * dz;
    unsigned m = (unsigned)__ballot(d <= R2_);
    int c = __popc(m);
    if (sel < 0 && ln >= cnt && ln < cnt + c) {
      int r = ln - cnt;
      unsigned mm = m;
      for (int i = 0; i < r; i++) mm &= (mm - 1);
      sel = base + (__ffs(mm) - 1);
    }
    cnt += c;
    if (cnt >= K_) break;   // uniform across the wave
  }
  int first = __shfl(sel, 0, 32);
  if (first < 0) first = far;
  const int g = (sel < 0) ? first : sel;

  const size_t col = ((size_t)(b * S_ + s)) * K_ + ln;
  X0[(size_t)0 * NT_ + col] = xb[g] - cx;
  X0[(size_t)1 * NT_ + col] = xb[N_ + g] - cy;
  X0[(size_t)2 * NT_ + col] = xb[2 * N_ + g] - cz;
  const float* pb = points + (size_t)b * D_ * N_;
#pragma unroll 4
  for (int d = 0; d < D_; d++)
    X0[(size_t)(3 + d) * NT_ + col] = pb[(size_t)d * N_ + g];
  X0[(size_t)67 * NT_ + col] = 0.0f;   // zero pad row (cin 67 -> 68)
}

// ---------------- Pack weights with zero-padded cin ----------------
__global__ void vn_packw(const float* __restrict__ W, float* __restrict__ Wp,
                         int cout, int cin, int cinp) {
  int idx = blockIdx.x * blockDim.x + threadIdx.x;
  if (idx >= cout * cinp) return;
  int r = idx / cinp, c = idx % cinp;
  Wp[idx] = (c < cin) ? W[r * cin + c] : 0.0f;
}

// ---------------- GEMM: Y[cout, NT] = Wp[cout, CINP] @ X[CINP, NT] ------------
// fp32 WMMA 16x16x4. One wave -> 16(M) x 64(N): A panel preloaded to registers
// once, 4 independent accumulators so v_wmma issues back-to-back with loads in
// flight (no per-wmma loadcnt-0 stall, no WMMA->WMMA RAW hazard).
// Bias is skipped: it cancels exactly under the following batch-norm.
template <int CINP>
__global__ __launch_bounds__(256) void vn_gemmT(const float* __restrict__ Wp,
                                                const float* __restrict__ X,
                                                float* __restrict__ Y) {
  constexpr int KSTEPS = CINP / 4;
  const int wv = threadIdx.x >> 5;
  const int ln = threadIdx.x & 31;
  const int colbase = blockIdx.x * 512 + wv * 64;   // 8 waves * 64 cols
  const int mbase = blockIdx.y * 16;
  const int m = ln & 15;
  const int koff = (ln >> 4) << 1;                  // 0 (lanes 0-15) or 2 (16-31)

  // Preload the wave's whole A panel (16 x CINP) into registers.
  const float* wrow = Wp + (size_t)(mbase + m) * CINP;
  v2f a[KSTEPS];
#pragma unroll
  for (int k = 0; k < KSTEPS; k++) {
    a[k].x = wrow[4 * k + koff];
    a[k].y = wrow[4 * k + koff + 1];
  }

  const float* xcol = X + colbase + m;              // n == ln & 15
  v8f acc0 = {}, acc1 = {}, acc2 = {}, acc3 = {};
#pragma unroll
  for (int k = 0; k < KSTEPS; k++) {
    const float* xr0 = xcol + (size_t)(4 * k + koff) * NT_;
    const float* xr1 = xcol + (size_t)(4 * k + koff + 1) * NT_;
    v2f b0, b1, b2, b3;
    b0.x = xr0[0];  b0.y = xr1[0];
    b1.x = xr0[16]; b1.y = xr1[16];
    b2.x = xr0[32]; b2.y = xr1[32];
    b3.x = xr0[48]; b3.y = xr1[48];
    acc0 = __builtin_amdgcn_wmma_f32_16x16x4_f32(false, a[k], false, b0, (short)0, acc0, false, false);
    acc1 = __builtin_amdgcn_wmma_f32_16x16x4_f32(false, a[k], false, b1, (short)0, acc1, false, false);
    acc2 = __builtin_amdgcn_wmma_f32_16x16x4_f32(false, a[k], false, b2, (short)0, acc2, false, false);
    acc3 = __builtin_amdgcn_wmma_f32_16x16x4_f32(false, a[k], false, b3, (short)0, acc3, false, false);
  }

  const int half = ln >> 4;                         // C/D: VGPR r -> rows r / r+8
  float* ybase = Y + colbase + (ln & 15);
#pragma unroll
  for (int r = 0; r < 8; r++) {
    const size_t ro = (size_t)(mbase + r + half * 8) * NT_;
    ybase[ro + 0]  = acc0[r];
    ybase[ro + 16] = acc1[r];
    ybase[ro + 32] = acc2[r];
    ybase[ro + 48] = acc3[r];
  }
}

// ------- Deterministic per-channel mean/var -> scale/shift (one block/channel) -------
__global__ __launch_bounds__(256) void vn_stats(const float* __restrict__ Y,
                                                const float* __restrict__ g,
                                                const float* __restrict__ be,
                                                float* __restrict__ scale,
                                                float* __restrict__ shift) {
  const int ch = blockIdx.x;
  const int t = threadIdx.x;
  const float* row = Y + (size_t)ch * NT_;
  float s = 0.f, q = 0.f;
  for (int i = t; i < NT_; i += 256) { float v = row[i]; s += v; q += v * v; }
  __shared__ float ss[256], sq[256];
  ss[t] = s; sq[t] = q;
  __syncthreads();
  for (int off = 128; off; off >>= 1) {
    if (t < off) { ss[t] += ss[t + off]; sq[t] += sq[t + off]; }
    __syncthreads();
  }
  if (t == 0) {
    float mu = ss[0] / (float)NT_;
    float var = sq[0] / (float)NT_ - mu * mu;
    float sc = g[ch] * rsqrtf(var + EPS_);
    scale[ch] = sc;
    shift[ch] = be[ch] - mu * sc;
  }
}

// ---------------- BN apply + ReLU, in place (layers 0 and 1) ----------------
__global__ __launch_bounds__(256) void vn_bnrelu(float* __restrict__ Y,
                                                 const float* __restrict__ scale,
                                                 const float* __restrict__ shift) {
  const int ch = blockIdx.y;
  const size_t i = (size_t)ch * NT_ + (size_t)blockIdx.x * 256 + threadIdx.x;
  Y[i] = fmaxf(Y[i] * scale[ch] + shift[ch], 0.0f);
}

// ------- Layer 2: fused BN + ReLU + max over K=32 neighbors -> feat[B,256,S] -------
__global__ __launch_bounds__(256) void vn_bnmaxpool(const float* __restrict__ Y,
                                                    const float* __restrict__ scale,
                                                    const float* __restrict__ shift,
                                                    float* __restrict__ feat) {
  int tid = blockIdx.x * 256 + threadIdx.x;
  if (tid >= B_ * 256 * S_) return;
  int s = tid % S_;
  int o = (tid / S_) % 256;
  int b = tid / (256 * S_);
  const float sc = scale[o], sh = shift[o];
  const float* src = Y + (size_t)o * NT_ + ((size_t)(b * S_ + s)) * K_;
  float mx = 0.0f;  // relu floor
#pragma unroll
  for (int k = 0; k < K_; k++) mx = fmaxf(mx, src[k] * sc + sh);
  feat[tid] = mx;
}

extern "C" void kernel_launch(void* const* d_in, const int* in_sizes, int n_in,
                              void* d_out, int out_size, void* d_ws, size_t ws_size,
                              hipStream_t stream) {
  (void)in_sizes; (void)n_in; (void)out_size; (void)ws_size;
  const float* xyz       = (const float*)d_in[0];
  const float* points    = (const float*)d_in[1];
  const int*   seed_inds = (const int*)d_in[2];
  const int*   fps_start = (const int*)d_in[3];
  const float* W0 = (const float*)d_in[4];
  const float* g0 = (const float*)d_in[6];
  const float* be0 = (const float*)d_in[7];
  const float* W1 = (const float*)d_in[8];
  const float* g1 = (const float*)d_in[10];
  const float* be1 = (const float*)d_in[11];
  const float* W2 = (const float*)d_in[12];
  const float* g2 = (const float*)d_in[14];
  const float* be2 = (const float*)d_in[15];

  float* out      = (float*)d_out;
  float* out_nx   = out;                                  // B*3*S
  float* out_feat = out + B_ * 3 * S_;                    // B*256*S
  float* out_seed = out_feat + (size_t)B_ * 256 * S_;     // B*S

  float* ws     = (float*)d_ws;
  int*   fpsIdx = (int*)ws;                               // B*S ints
  float* Wp0    = ws + B_ * S_;                           // 128*68
  float* Wp1    = Wp0 + 128 * 68;                         // 128*128
  float* Wp2    = Wp1 + 128 * 128;                        // 256*128
  float* scale  = Wp2 + 256 * 128;                        // 256
  float* shift  = scale + 256;                            // 256
  float* bufA   = shift + 256 + 256;                      // 256*NT floats
  float* bufB   = bufA + (size_t)256 * NT_;               // 256*NT floats

  vn_fps<<<B_, 1024, 0, stream>>>(xyz, fps_start, fpsIdx);
  vn_packw<<<(128 * 68 + 255) / 256, 256, 0, stream>>>(W0, Wp0, 128, 67, 68);
  vn_packw<<<(128 * 128 + 255) / 256, 256, 0, stream>>>(W1, Wp1, 128, 128, 128);
  vn_packw<<<(256 * 128 + 255) / 256, 256, 0, stream>>>(W2, Wp2, 256, 128, 128);
  vn_ballgather<<<B_ * S_ / 8, 256, 0, stream>>>(xyz, points, seed_inds, fpsIdx,
                                                 bufA, out_nx, out_seed);
  // layer 0: 68 -> 128   (bufA -> bufB)
  vn_gemmT<68><<<dim3(NT_ / 512, 128 / 16), 256, 0, stream>>>(Wp0, bufA, bufB);
  vn_stats<<<128, 256, 0, stream>>>(bufB, g0, be0, scale, shift);
  vn_bnrelu<<<dim3(NT_ / 256, 128), 256, 0, stream>>>(bufB, scale, shift);
  // layer 1: 128 -> 128  (bufB -> bufA)
  vn_gemmT<128><<<dim3(NT_ / 512, 128 / 16), 256, 0, stream>>>(Wp1, bufB, bufA);
  vn_stats<<<128, 256, 0, stream>>>(bufA, g1, be1, scale, shift);
  vn_bnrelu<<<dim3(NT_ / 256, 128), 256, 0, stream>>>(bufA, scale, shift);
  // layer 2: 128 -> 256  (bufA -> bufB), BN+ReLU fused into maxpool
  vn_gemmT<128><<<dim3(NT_ / 512, 256 / 16), 256, 0, stream>>>(Wp2, bufA, bufB);
  vn_stats<<<256, 256, 0, stream>>>(bufB, g2, be2, scale, shift);
  vn_bnmaxpool<<<(B_ * 256 * S_ + 255) / 256, 256, 0, stream>>>(bufB, scale, shift,
                                                                out_feat);
}